// Loss_38285338476968
// MI455X (gfx1250) — compile-verified
//
#include <hip/hip_runtime.h>

typedef __attribute__((ext_vector_type(2))) float v2f;
typedef __attribute__((ext_vector_type(8))) float v8f;

#define HW         4096    // 64*64 elements per batch
#define BPW        16      // batches per workgroup (WMMA M tile)
#define CHUNK      512     // K elements staged per chunk
#define NCHUNK     (HW / CHUNK)   // 8
#define LDS_STRIDE 516     // 512 + 4 pad -> conflict-free WMMA A reads

__global__ __launch_bounds__(256)
void moments_argmax_kernel(const float* __restrict__ t, float* __restrict__ blockOut) {
    __shared__ float ldsA[BPW * LDS_STRIDE];   // 33,024 B staging tile
    __shared__ float momLds[8 * BPW * 4];      // per-wave moment stash
    __shared__ float redV[256];
    __shared__ int   redI[256];
    __shared__ float mom[BPW * 4];
    __shared__ int   amax[BPW];
    __shared__ float lossLds[BPW];

    const int tid  = threadIdx.x;
    const int wave = tid >> 5;
    const int lane = tid & 31;
    const int b0   = blockIdx.x * BPW;

    // phase-A (load/argmax) assignment: 16 threads per batch
    const int am = tid >> 4;   // batch row within tile, 0..15
    const int ap = tid & 15;   // sub-lane within batch

    // phase-B (WMMA) lane constants
    const int brow = lane & 15;               // A-matrix row (batch)
    const int hi   = (lane >= 16) ? 2 : 0;    // K sub-offset for upper half-wave
    const int ncol = lane & 15;               // B/D column (moment index)

    float maxv = -1.0f;   // inputs are uniform in [0,1)
    int   maxi = 0;

    v8f acc = {};         // WMMA accumulator: D[m][n] moments

    const float* tb = t + (size_t)b0 * HW;

    for (int c = 0; c < NCHUNK; ++c) {
        const int kb = c * CHUNK;

        // ---- Phase A: coalesced load, argmax scan, stage to LDS ----
        #pragma unroll
        for (int it = 0; it < 8; ++it) {
            const int klocal = it * 64 + ap * 4;
            const int kabs   = kb + klocal;
            const float4 v = *(const float4*)(tb + (size_t)am * HW + kabs);
            if (v.x > maxv) { maxv = v.x; maxi = kabs; }
            if (v.y > maxv) { maxv = v.y; maxi = kabs + 1; }
            if (v.z > maxv) { maxv = v.z; maxi = kabs + 2; }
            if (v.w > maxv) { maxv = v.w; maxi = kabs + 3; }
            *(float4*)&ldsA[am * LDS_STRIDE + klocal] = v;
        }
        if (c + 1 < NCHUNK) {
            __builtin_prefetch(tb + (size_t)am * HW + kb + CHUNK + ap * 4, 0, 0);
        }
        __syncthreads();

        // ---- Phase B: each wave accumulates its 64-wide K slice via WMMA ----
        #pragma unroll
        for (int s = 0; s < 16; ++s) {
            const int kk = wave * 64 + s * 4;   // local K base within chunk
            const int f  = kb + kk;             // absolute flat index base

            // A fragment: t[batch brow][f + hi], t[batch brow][f + hi + 1]
            v2f a;
            a.x = ldsA[brow * LDS_STRIDE + kk + hi];
            a.y = ldsA[brow * LDS_STRIDE + kk + hi + 1];

            // B fragment: weight w(i, ncol) for i = f+hi, f+hi+1
            const int i0 = f + hi;
            const int j0 = i0 >> 6, k0 = i0 & 63;
            const int i1 = i0 + 1;
            const int j1 = i1 >> 6, k1 = i1 & 63;
            float w0, w1;
            if      (ncol == 0) { w0 = 1.0f;        w1 = 1.0f; }
            else if (ncol == 1) { w0 = (float)j0;   w1 = (float)j1; }
            else if (ncol == 2) { w0 = (float)k0;   w1 = (float)k1; }
            else if (ncol == 3) { w0 = (float)(j0*j0 + k0*k0);
                                  w1 = (float)(j1*j1 + k1*k1); }
            else                { w0 = 0.0f;        w1 = 0.0f; }
            v2f b; b.x = w0; b.y = w1;

            acc = __builtin_amdgcn_wmma_f32_16x16x4_f32(
                      false, a, false, b, (short)0, acc, false, false);
        }
        __syncthreads();
    }

    // ---- stash per-wave WMMA accumulators (cols 0..3 are live) ----
    if (ncol < 4) {
        #pragma unroll
        for (int r = 0; r < 8; ++r) {
            const int row = r + ((lane >= 16) ? 8 : 0);
            momLds[(wave * BPW + row) * 4 + ncol] = acc[r];
        }
    }
    redV[tid] = maxv;
    redI[tid] = maxi;
    __syncthreads();

    // cross-wave moment reduction: fixed order
    if (tid < BPW * 4) {
        const int m = tid >> 2, n = tid & 3;
        float s = 0.0f;
        for (int w = 0; w < 8; ++w) s += momLds[(w * BPW + m) * 4 + n];
        mom[tid] = s;
    }
    // per-batch argmax reduction with explicit min-index tie-break
    if (tid < BPW) {
        float bv = redV[tid * 16];
        int   bi = redI[tid * 16];
        for (int p = 1; p < 16; ++p) {
            const float v = redV[tid * 16 + p];
            const int   i = redI[tid * 16 + p];
            if (v > bv || (v == bv && i < bi)) { bv = v; bi = i; }
        }
        amax[tid] = bi;
    }
    __syncthreads();

    if (tid < BPW) {
        const float S  = mom[tid * 4 + 0];
        const float Rj = mom[tid * 4 + 1];
        const float Ck = mom[tid * 4 + 2];
        const float Q  = mom[tid * 4 + 3];
        const int   mi = amax[tid];
        const float mx = (float)(mi >> 6);
        const float my = (float)(mi & 63);
        lossLds[tid] = (mx * mx + my * my) * S - 2.0f * mx * Rj - 2.0f * my * Ck + Q;
    }
    __syncthreads();

    if (tid == 0) {
        float s = 0.0f;
        for (int i = 0; i < BPW; ++i) s += lossLds[i];
        blockOut[blockIdx.x] = s;
    }
}

__global__ __launch_bounds__(256)
void finalize_kernel(const float* __restrict__ part, float* __restrict__ out, int n) {
    __shared__ double red[256];
    const int tid = threadIdx.x;
    double s = 0.0;
    for (int i = tid; i < n; i += 256) s += (double)part[i];
    red[tid] = s;
    __syncthreads();
    for (int off = 128; off > 0; off >>= 1) {
        if (tid < off) red[tid] += red[tid + off];
        __syncthreads();
    }
    if (tid == 0) out[0] = (float)red[0];
}

extern "C" void kernel_launch(void* const* d_in, const int* in_sizes, int n_in,
                              void* d_out, int out_size, void* d_ws, size_t ws_size,
                              hipStream_t stream) {
    const float* t = (const float*)d_in[0];
    const int total   = in_sizes[0];        // 8192 * 64 * 64
    const int nbatch  = total / HW;         // 8192
    const int nblocks = nbatch / BPW;       // 512

    float* part = (float*)d_ws;             // 512 floats of scratch

    moments_argmax_kernel<<<nblocks, 256, 0, stream>>>(t, part);
    finalize_kernel<<<1, 256, 0, stream>>>(part, (float*)d_out, nblocks);
}